// HeatmapProcessor2_70385924047590
// MI455X (gfx1250) — compile-verified
//
#include <hip/hip_runtime.h>
#include <math.h>

typedef float v2f __attribute__((ext_vector_type(2)));
typedef float v8f __attribute__((ext_vector_type(8)));

#define IN_H   64
#define IN_W   32
#define OUT_H  16
#define OUT_W  8
#define NCH    17
#define NGRP   3
#define HW     (OUT_H * OUT_W)   /* 128 */
#define NWAVES 8

// Antialiased linear-resize weight (jax.image.resize 'linear', scale = 1/4,
// half-pixel centers). Triangle kernel scaled by 1/4; output o, input j.
// Interior rows normalize by 4, boundary rows (2 clipped taps) by 3.5.
__device__ __forceinline__ float resize_w(int o, int j, int osz) {
    float c = 4.0f * (float)o + 1.5f;
    float t = 1.0f - 0.25f * fabsf((float)j - c);
    t = fmaxf(t, 0.0f);
    float norm = (o == 0 || o == osz - 1) ? 3.5f : 4.0f;
    return t / norm;
}

__global__ __launch_bounds__(256)
void pose_heatmap_kernel(const float* __restrict__ x,
                         float* __restrict__ out_hm,   // [N,3,16,8]
                         float* __restrict__ out_mr,   // [N,3]
                         float* __restrict__ out_mi) { // [N,17,2]
    __shared__ float sWh[OUT_H * IN_H];          // 16x64 row-resize weights
    __shared__ float sWv[OUT_W * IN_W];          // 8x32 col-resize weights
    __shared__ float sGrp[NGRP * HW];            // group-summed heatmaps
    __shared__ float sChMax[NCH];                // per-channel max response
    __shared__ float sR[NWAVES][OUT_H * IN_W];   // per-wave 16x32 intermediate

    const int tid  = threadIdx.x;
    const int n    = blockIdx.x;
    const int lane = tid & 31;
    const int wave = tid >> 5;
    const int h16  = (lane >> 4) & 1;   // lane half (0: lanes 0-15, 1: 16-31)
    const int lm   = lane & 15;

    // ---- phase 0: build weights, zero group accumulators ----
    for (int i = tid; i < OUT_H * IN_H; i += 256) sWh[i] = resize_w(i / IN_H, i % IN_H, OUT_H);
    for (int i = tid; i < OUT_W * IN_W; i += 256) sWv[i] = resize_w(i / IN_W, i % IN_W, OUT_W);
    for (int i = tid; i < NGRP * HW;    i += 256) sGrp[i] = 0.0f;
    __syncthreads();

    // A fragments of Wh (16x4 f32 layout: lane = M, K = 2*half + vgpr)
    v2f aW[16];
#pragma unroll
    for (int kk = 0; kk < 16; ++kk) {
        aW[kk].x = sWh[lm * IN_H + 4 * kk + 2 * h16 + 0];
        aW[kk].y = sWh[lm * IN_H + 4 * kk + 2 * h16 + 1];
    }
    // B fragments of Wv^T (4x16, N padded 8->16 with zeros)
    v2f bW[8];
#pragma unroll
    for (int kk = 0; kk < 8; ++kk) {
        if (lm < OUT_W) {
            bW[kk].x = sWv[lm * IN_W + 4 * kk + 2 * h16 + 0];
            bW[kk].y = sWv[lm * IN_W + 4 * kk + 2 * h16 + 1];
        } else {
            bW[kk].x = 0.0f; bW[kk].y = 0.0f;
        }
    }

    float* myR = &sR[wave][0];

    // ---- phase 1: per-channel resize (WMMA) + max/argmax + group sums ----
    for (int c = wave; c < NCH; c += NWAVES) {
        const float* xc = x + ((size_t)n * NCH + c) * (IN_H * IN_W);
        __builtin_prefetch(xc + lane * 64, 0, 0);   // global_prefetch_b8, warm L2

        // Row pass: R(16x32) = Wh(16x64) * X(64x32), two 16-col tiles, K=64.
        v8f acc0 = {}; v8f acc1 = {};
#pragma unroll
        for (int kk = 0; kk < 16; ++kk) {
            int r = 4 * kk + 2 * h16;
            v2f b0, b1;            // B 4x16 tile: [K = 2*half + vgpr][N = lm]
            b0.x = xc[(r + 0) * IN_W + lm];
            b0.y = xc[(r + 1) * IN_W + lm];
            b1.x = xc[(r + 0) * IN_W + 16 + lm];
            b1.y = xc[(r + 1) * IN_W + 16 + lm];
            acc0 = __builtin_amdgcn_wmma_f32_16x16x4_f32(false, aW[kk], false, b0,
                                                         (short)0, acc0, false, false);
            acc1 = __builtin_amdgcn_wmma_f32_16x16x4_f32(false, aW[kk], false, b1,
                                                         (short)0, acc1, false, false);
        }
        // Stage R in LDS from C/D layout (row M = v + 8*half, col = tile*16 + lm).
#pragma unroll
        for (int v = 0; v < 8; ++v) {
            myR[(v + 8 * h16) * IN_W + lm]      = acc0[v];
            myR[(v + 8 * h16) * IN_W + 16 + lm] = acc1[v];
        }
        // Column pass: OUT(16x8, padded to 16 cols) = R(16x32) * Wv^T(32x8), K=32.
        v8f acc2 = {};
#pragma unroll
        for (int kk = 0; kk < 8; ++kk) {
            v2f aR;
            aR.x = myR[lm * IN_W + 4 * kk + 2 * h16 + 0];
            aR.y = myR[lm * IN_W + 4 * kk + 2 * h16 + 1];
            acc2 = __builtin_amdgcn_wmma_f32_16x16x4_f32(false, aR, false, bW[kk],
                                                         (short)0, acc2, false, false);
        }

        // Per-channel max/argmax over the 16x8 tile (first-occurrence ties).
        float bv = -3.402823466e38f; int bi = 0;
        if (lm < OUT_W) {
#pragma unroll
            for (int v = 0; v < 8; ++v) {
                int   idx = (v + 8 * h16) * OUT_W + lm;
                float val = acc2[v];
                if (val > bv || (val == bv && idx < bi)) { bv = val; bi = idx; }
            }
        }
#pragma unroll
        for (int m = 16; m >= 1; m >>= 1) {       // wave32 reduction
            float ov = __shfl_xor(bv, m, 32);
            int   oi = __shfl_xor(bi, m, 32);
            if (ov > bv || (ov == bv && oi < bi)) { bv = ov; bi = oi; }
        }

        const int g = (c < 5) ? 0 : (c < 11 ? 1 : 2);
        if (lm < OUT_W) {
#pragma unroll
            for (int v = 0; v < 8; ++v)
                atomicAdd(&sGrp[g * HW + (v + 8 * h16) * OUT_W + lm], acc2[v]); // ds_add_f32
        }
        if (lane == 0) {
            sChMax[c] = bv;
            out_mi[((size_t)n * NCH + c) * 2 + 0] = (float)(bi % OUT_W);
            out_mi[((size_t)n * NCH + c) * 2 + 1] = (float)(bi / OUT_W);
        }
    }
    __syncthreads();

    // ---- phase 2: softmax per group + group mean responses (waves 0..2) ----
    if (wave < NGRP) {
        const int g = wave;
        float v0 = sGrp[g * HW + lane];
        float v1 = sGrp[g * HW + lane + 32];
        float v2 = sGrp[g * HW + lane + 64];
        float v3 = sGrp[g * HW + lane + 96];
        float mx = fmaxf(fmaxf(v0, v1), fmaxf(v2, v3));
#pragma unroll
        for (int m = 16; m >= 1; m >>= 1) mx = fmaxf(mx, __shfl_xor(mx, m, 32));
        float e0 = __expf(v0 - mx), e1 = __expf(v1 - mx);
        float e2 = __expf(v2 - mx), e3 = __expf(v3 - mx);
        float s = e0 + e1 + e2 + e3;
#pragma unroll
        for (int m = 16; m >= 1; m >>= 1) s += __shfl_xor(s, m, 32);
        float inv = 1.0f / s;
        float* o = out_hm + ((size_t)n * NGRP + g) * HW;
        o[lane]      = e0 * inv;
        o[lane + 32] = e1 * inv;
        o[lane + 64] = e2 * inv;
        o[lane + 96] = e3 * inv;
        if (lane == 0) {
            const int st  = (g == 0) ? 0 : (g == 1 ? 5 : 11);
            const int cnt = (g == 0) ? 5 : 6;
            float sum = 0.0f;
            for (int k = 0; k < cnt; ++k) sum += sChMax[st + k];
            out_mr[(size_t)n * NGRP + g] = sum / (float)cnt;
        }
    }
}

extern "C" void kernel_launch(void* const* d_in, const int* in_sizes, int n_in,
                              void* d_out, int out_size, void* d_ws, size_t ws_size,
                              hipStream_t stream) {
    const float* x = (const float*)d_in[0];
    float* out = (float*)d_out;
    const int N = in_sizes[0] / (NCH * IN_H * IN_W);   // 2048

    float* out_hm = out;                              // N*3*128 heatmaps
    float* out_mr = out + (size_t)N * NGRP * HW;      // N*3 mean max responses
    float* out_mi = out_mr + (size_t)N * NGRP;        // N*17*2 argmax coords

    hipLaunchKernelGGL(pose_heatmap_kernel, dim3(N), dim3(256), 0, stream,
                       x, out_hm, out_mr, out_mi);
}